// irnn_layer_15187004358988
// MI455X (gfx1250) — compile-verified
//
#include <hip/hip_runtime.h>

#ifndef __has_builtin
#define __has_builtin(x) 0
#endif

#if __has_builtin(__builtin_amdgcn_global_load_async_to_lds_b128) && \
    __has_builtin(__builtin_amdgcn_s_wait_asynccnt)
#define HAVE_ASYNC_LDS 1
#else
#define HAVE_ASYNC_LDS 0
#endif

typedef float v4f __attribute__((ext_vector_type(4)));

namespace {
constexpr int  kB = 8, kC = 32, kH = 256, kW = 256;
constexpr long long kPlane = (long long)kB * kC * kH * kW;   // 16,777,216
constexpr int  kRowsPerBlock = 256;   // hscan: rows handled per block (== kH)
constexpr int  kChunk = 32;           // columns per LDS chunk
constexpr int  kPad   = 36;           // padded LDS row stride; 36*4B = 144B -> 16B aligned
constexpr int  kNChunks = kW / kChunk;          // 8
}

#if HAVE_ASYNC_LDS
typedef int v4i __attribute__((vector_size(16)));                 // int4, gcc-style
typedef __attribute__((address_space(1))) v4i GInt4;
typedef __attribute__((address_space(3))) v4i LInt4;
#endif

// ---------------------------------------------------------------------------
// Vertical scans (top_down, top_up): one thread per (b,c,w) column.
// Consecutive lanes -> consecutive w -> fully coalesced 128B/wave accesses.
// Prefetch hoisted to once per 64-row segment.
// ---------------------------------------------------------------------------
__global__ __launch_bounds__(256) void vscan_kernel(
    const float* __restrict__ x,
    const float* __restrict__ w_up, const float* __restrict__ b_up,
    const float* __restrict__ w_dn, const float* __restrict__ b_dn,
    float* __restrict__ out_up, float* __restrict__ out_dn)
{
  const int gid = blockIdx.x * blockDim.x + threadIdx.x;   // [0, B*C*W)
  const int w   = gid & (kW - 1);
  const int bc  = gid >> 8;                                // kW == 256
  const int c   = bc & (kC - 1);                           // wave-uniform -> scalar loads
  const size_t base = (size_t)bc * kH * kW + (size_t)w;

  const float wd = w_dn[c], bd = b_dn[c];
  const float wu = w_up[c], bu = b_up[c];

  // ---- top-down: h = 0 .. H-1 ----
  {
    float h = x[base];
    __builtin_nontemporal_store(h, out_dn + base);
    #pragma unroll 1
    for (int seg = 0; seg < kH / 64; ++seg) {
      int pf = seg * 64 + 96; if (pf > kH - 1) pf = kH - 1;
      __builtin_prefetch(x + base + (size_t)pf * kW, 0, 0);   // global_prefetch_b8
      const int t0 = (seg == 0) ? 1 : seg * 64;
      const int t1 = seg * 64 + 64;
      #pragma unroll 4
      for (int t = t0; t < t1; ++t) {
        const float v = x[base + (size_t)t * kW];
        h = fmaxf(0.0f, __fmaf_rn(wd, h, bd) + v);
        __builtin_nontemporal_store(h, out_dn + base + (size_t)t * kW);
      }
    }
  }
  // ---- bottom-up: h = H-1 .. 0 ----
  {
    float h = x[base + (size_t)(kH - 1) * kW];
    __builtin_nontemporal_store(h, out_up + base + (size_t)(kH - 1) * kW);
    #pragma unroll 1
    for (int seg = kH / 64 - 1; seg >= 0; --seg) {
      int pf = seg * 64 - 32; if (pf < 0) pf = 0;
      __builtin_prefetch(x + base + (size_t)pf * kW, 0, 0);
      const int t1 = (seg == kH / 64 - 1) ? kH - 2 : seg * 64 + 63;
      const int t0 = seg * 64;
      #pragma unroll 4
      for (int t = t1; t >= t0; --t) {
        const float v = x[base + (size_t)t * kW];
        h = fmaxf(0.0f, __fmaf_rn(wu, h, bu) + v);
        __builtin_nontemporal_store(h, out_up + base + (size_t)t * kW);
      }
    }
  }
}

// ---------------------------------------------------------------------------
// Horizontal scans (top_right, top_left). W is contiguous, so stage 256x32
// tiles through padded LDS. All paths are 128-bit wide:
//   fill : 8x global_load_async_to_lds_b128 per wave per chunk
//   scan : per-thread row as 8x float4 from LDS (16B-aligned, stride 36)
//   store: 8x b128 non-temporal stores per wave per chunk
// All 256 rows of a block share one channel -> scalar weights.
// ---------------------------------------------------------------------------
__global__ __launch_bounds__(256) void hscan_kernel(
    const float* __restrict__ x,
    const float* __restrict__ w_rt, const float* __restrict__ b_rt,
    const float* __restrict__ w_lt, const float* __restrict__ b_lt,
    float* __restrict__ out_rt, float* __restrict__ out_lt)
{
  __shared__ __align__(16) float tile[kRowsPerBlock * kPad];   // 36,864 B

  const int tid  = threadIdx.x;
  const int lane = tid & 31;
  const int wv   = tid >> 5;                     // wave id 0..7
  const int sub  = lane & 7;                     // 16B column group within chunk
  const int grp  = lane >> 3;                    // row within group-of-4
  const size_t rowBase = (size_t)blockIdx.x * kRowsPerBlock;
  const int c = blockIdx.x & (kC - 1);           // rows-per-(b,c) == kH == kRowsPerBlock

  const float wr = w_rt[c], br = b_rt[c];
  const float wl = w_lt[c], bl = b_lt[c];
  v4f* const row4 = (v4f*)&tile[tid * kPad];     // this thread's scan row

  // ---------------- RIGHT scan: chunks left -> right ----------------
  float hR = 0.0f;
  for (int k = 0; k < kNChunks; ++k) {
    const int c0 = k * kChunk;
    // fill: wave loads its own 32 rows x 32 cols, 16B per lane
    #pragma unroll
    for (int i = 0; i < 8; ++i) {
      const int r = (wv << 5) + (i << 2) + grp;
      const float* gp = x + (rowBase + (size_t)r) * kW + (size_t)(c0 + (sub << 2));
      float* lp = &tile[r * kPad + (sub << 2)];
#if HAVE_ASYNC_LDS
      __builtin_amdgcn_global_load_async_to_lds_b128(
          (GInt4*)(float*)gp, (LInt4*)lp, 0, 0);
#else
      *(v4f*)lp = *(const v4f*)gp;
#endif
    }
#if HAVE_ASYNC_LDS
    __builtin_amdgcn_s_wait_asynccnt(0);
#endif
    __syncthreads();

    // scan this thread's row, 4 elements per LDS vector, in place
    #pragma unroll
    for (int jv = 0; jv < kChunk / 4; ++jv) {
      v4f v = row4[jv];
      if (k == 0 && jv == 0) hR = v.x;
      else hR = fmaxf(0.0f, __fmaf_rn(wr, hR, br) + v.x);
      v.x = hR;
      hR = fmaxf(0.0f, __fmaf_rn(wr, hR, br) + v.y); v.y = hR;
      hR = fmaxf(0.0f, __fmaf_rn(wr, hR, br) + v.z); v.z = hR;
      hR = fmaxf(0.0f, __fmaf_rn(wr, hR, br) + v.w); v.w = hR;
      row4[jv] = v;
    }
    __syncthreads();

    // store: mirror of fill, non-temporal b128
    #pragma unroll
    for (int i = 0; i < 8; ++i) {
      const int r = (wv << 5) + (i << 2) + grp;
      const v4f v = *(const v4f*)&tile[r * kPad + (sub << 2)];
      __builtin_nontemporal_store(
          v, (v4f*)(out_rt + (rowBase + (size_t)r) * kW + (size_t)(c0 + (sub << 2))));
    }
    __syncthreads();                             // WAR before next chunk's fill
  }

  // ---------------- LEFT scan: chunks right -> left ----------------
  float hL = 0.0f;
  for (int k = kNChunks - 1; k >= 0; --k) {
    const int c0 = k * kChunk;
    #pragma unroll
    for (int i = 0; i < 8; ++i) {
      const int r = (wv << 5) + (i << 2) + grp;
      const float* gp = x + (rowBase + (size_t)r) * kW + (size_t)(c0 + (sub << 2));
      float* lp = &tile[r * kPad + (sub << 2)];
#if HAVE_ASYNC_LDS
      __builtin_amdgcn_global_load_async_to_lds_b128(
          (GInt4*)(float*)gp, (LInt4*)lp, 0, 0);
#else
      *(v4f*)lp = *(const v4f*)gp;
#endif
    }
#if HAVE_ASYNC_LDS
    __builtin_amdgcn_s_wait_asynccnt(0);
#endif
    __syncthreads();

    #pragma unroll
    for (int jv = kChunk / 4 - 1; jv >= 0; --jv) {
      v4f v = row4[jv];
      if (k == kNChunks - 1 && jv == kChunk / 4 - 1) hL = v.w;
      else hL = fmaxf(0.0f, __fmaf_rn(wl, hL, bl) + v.w);
      v.w = hL;
      hL = fmaxf(0.0f, __fmaf_rn(wl, hL, bl) + v.z); v.z = hL;
      hL = fmaxf(0.0f, __fmaf_rn(wl, hL, bl) + v.y); v.y = hL;
      hL = fmaxf(0.0f, __fmaf_rn(wl, hL, bl) + v.x); v.x = hL;
      row4[jv] = v;
    }
    __syncthreads();

    #pragma unroll
    for (int i = 0; i < 8; ++i) {
      const int r = (wv << 5) + (i << 2) + grp;
      const v4f v = *(const v4f*)&tile[r * kPad + (sub << 2)];
      __builtin_nontemporal_store(
          v, (v4f*)(out_lt + (rowBase + (size_t)r) * kW + (size_t)(c0 + (sub << 2))));
    }
    __syncthreads();
  }
}

// ---------------------------------------------------------------------------
extern "C" void kernel_launch(void* const* d_in, const int* in_sizes, int n_in,
                              void* d_out, int out_size, void* d_ws, size_t ws_size,
                              hipStream_t stream)
{
  (void)in_sizes; (void)n_in; (void)out_size; (void)d_ws; (void)ws_size;

  const float* x    = (const float*)d_in[0];
  const float* w_up = (const float*)d_in[1];
  const float* b_up = (const float*)d_in[2];
  const float* w_rt = (const float*)d_in[3];
  const float* b_rt = (const float*)d_in[4];
  const float* w_dn = (const float*)d_in[5];
  const float* b_dn = (const float*)d_in[6];
  const float* w_lt = (const float*)d_in[7];
  const float* b_lt = (const float*)d_in[8];

  float* out    = (float*)d_out;          // order: up, right, down, left
  float* out_up = out;
  float* out_rt = out + kPlane;
  float* out_dn = out + 2 * kPlane;
  float* out_lt = out + 3 * kPlane;

  // Vertical: one thread per (b,c,w) column -> 65536 threads (2048 waves).
  vscan_kernel<<<(kB * kC * kW) / 256, 256, 0, stream>>>(
      x, w_up, b_up, w_dn, b_dn, out_up, out_dn);

  // Horizontal: one block per 256 rows -> 256 blocks of 256 threads.
  hscan_kernel<<<(kB * kC * kH) / kRowsPerBlock, 256, 0, stream>>>(
      x, w_rt, b_rt, w_lt, b_lt, out_rt, out_lt);
}